// ConvQuadInterp3d_54460185313460
// MI455X (gfx1250) — compile-verified
//
#include <hip/hip_runtime.h>
#include <cstdint>

#define B_   4
#define D_   8
#define H_   384
#define W_   384
#define TW   64
#define TH   8
#define LH   (TH + 2)            // 10 rows (with h halo)
#define LROW 72                  // padded row: col 3 = left halo, 4..67 interior, 68 = right halo
#define LPL  (LH * LROW)         // 720 floats per depth plane
#define LN   (D_ * LPL)          // 5760 floats = 23040 B LDS
#define NPB  ((size_t)D_ * H_ * W_)              // per-batch voxels
#define CO   ((size_t)B_ * 3 * D_ * H_ * W_)     // coords element count

// ---------------------------------------------------------------------------
// Compile-time reproduction of jax.random.uniform(jax.random.key(1234),(3,3))
// via threefry2x32 (exact JAX schedule: counts iota(9), odd-size zero pad,
// x0=[0..4], x1=[5,6,7,8,0], 20 rounds, ks2 = k0^k1^0x1BD11BDA).
// ---------------------------------------------------------------------------
struct Jit { float v[9]; };

constexpr Jit make_jitter() {
  unsigned bits[10] = {};
  const unsigned k0 = 0u, k1 = 1234u;
  const unsigned k2 = k0 ^ k1 ^ 0x1BD11BDAu;
  const int RA[4] = {13, 15, 26, 6};
  const int RB[4] = {17, 29, 16, 24};
  for (int i = 0; i < 5; ++i) {
    unsigned x0 = (unsigned)i;
    unsigned x1 = (i < 4) ? (unsigned)(5 + i) : 0u;
    x0 += k0; x1 += k1;
    for (int r = 0; r < 4; ++r) { x0 += x1; x1 = (x1 << RA[r]) | (x1 >> (32 - RA[r])); x1 ^= x0; }
    x0 += k1; x1 += k2 + 1u;
    for (int r = 0; r < 4; ++r) { x0 += x1; x1 = (x1 << RB[r]) | (x1 >> (32 - RB[r])); x1 ^= x0; }
    x0 += k2; x1 += k0 + 2u;
    for (int r = 0; r < 4; ++r) { x0 += x1; x1 = (x1 << RA[r]) | (x1 >> (32 - RA[r])); x1 ^= x0; }
    x0 += k0; x1 += k1 + 3u;
    for (int r = 0; r < 4; ++r) { x0 += x1; x1 = (x1 << RB[r]) | (x1 >> (32 - RB[r])); x1 ^= x0; }
    x0 += k1; x1 += k2 + 4u;
    for (int r = 0; r < 4; ++r) { x0 += x1; x1 = (x1 << RA[r]) | (x1 >> (32 - RA[r])); x1 ^= x0; }
    x0 += k2; x1 += k0 + 5u;
    bits[i] = x0; bits[5 + i] = x1;
  }
  Jit j{};
  for (int i = 0; i < 9; ++i) {
    unsigned u = (bits[i] >> 9) | 0x3f800000u;                 // [1,2)
    j.v[i] = (__builtin_bit_cast(float, u) - 1.0f) * 1e-7f;    // uniform*EPS
  }
  return j;
}
constexpr Jit JIT = make_jitter();

__device__ __forceinline__ int clampi(int v, int lo, int hi) {
  return v < lo ? lo : (v > hi ? hi : v);
}

__global__ __launch_bounds__(256) void
ConvQuadInterp3d_kernel(const float* __restrict__ x, float* __restrict__ out) {
  __shared__ float tile[LN];   // [d][hh(10)][col(72)]; cols 3..68 valid

  const int tid = threadIdx.x;
  const int w0  = blockIdx.x * TW;
  const int h0  = blockIdx.y * TH;
  const int b   = blockIdx.z;
  const float* xb = x + (size_t)b * NPB;

  // ------------------------------------------------------------------
  // Stage the replicate-padded (8 x 10 x 66) tile into LDS via the
  // CDNA5 async global->LDS path (ASYNCcnt). Waves 0-4 (lanes 0-159):
  // 10 rows x 16 aligned b128 chunks covering the 64-wide interior.
  // Wave 5 (lanes 160-179): the 20 halo columns with b32. No divisions,
  // constant per-plane strides (global += H*W, LDS += plane).
  // ------------------------------------------------------------------
  if (tid < 160) {
    const int hh = tid >> 4;            // 0..9
    const int c  = (tid & 15) << 2;     // 0,4,...,60
    const int hg = clampi(h0 + hh - 1, 0, H_ - 1);
    const float* g = xb + (size_t)hg * W_ + (w0 + c);          // 16B aligned
    unsigned lds = (unsigned)(unsigned long long)(const void*)&tile[hh * LROW + 4 + c];
#pragma unroll
    for (int d = 0; d < D_; ++d) {
      asm volatile("global_load_async_to_lds_b128 %0, %1, off"
                   :: "v"(lds), "v"(g) : "memory");
      g   += (size_t)H_ * W_;
      lds += LPL * 4;
    }
  } else if (tid < 180) {
    const int idx  = tid - 160;
    const int side = idx >= 10;         // 0 = left halo, 1 = right halo
    const int hh   = side ? idx - 10 : idx;
    const int hg   = clampi(h0 + hh - 1, 0, H_ - 1);
    const int wg   = side ? (w0 + TW < W_ ? w0 + TW : W_ - 1)
                          : (w0 > 0 ? w0 - 1 : 0);
    const int col  = side ? (TW + 4) : 3;
    const float* g = xb + (size_t)hg * W_ + wg;
    unsigned lds = (unsigned)(unsigned long long)(const void*)&tile[hh * LROW + col];
#pragma unroll
    for (int d = 0; d < D_; ++d) {
      asm volatile("global_load_async_to_lds_b32 %0, %1, off"
                   :: "v"(lds), "v"(g) : "memory");
      g   += (size_t)H_ * W_;
      lds += LPL * 4;
    }
  }
  asm volatile("s_wait_asynccnt 0x0" ::: "memory");
  __syncthreads();

  // ---- compute: 2 columns per thread, rolling 3-plane (27-value) window ----
  for (int c = tid; c < TW * TH; c += 256) {
    const int wl = c & (TW - 1);
    const int hl = c >> 6;

    float pm[9], pc[9], pp[9];
    auto ldp = [&](int d, float* p) {
      const int base = (d * LH + hl) * LROW + wl + 3;
#pragma unroll
      for (int dh = 0; dh < 3; ++dh)
#pragma unroll
        for (int dw = 0; dw < 3; ++dw)
          p[dh * 3 + dw] = tile[base + dh * LROW + dw];
    };
    ldp(0, pc);
    ldp(1, pp);
#pragma unroll
    for (int k = 0; k < 9; ++k) pm[k] = pc[k];   // replicate d = -1

    for (int d = 0; d < D_; ++d) {
      const float xc = pc[4];
      // gradients (kornia depth-flip: gz, dys, dxs signs)
      const float gx  = 0.5f * (pc[5] - pc[3]);
      const float gy  = 0.5f * (pc[7] - pc[1]);
      const float gz  = 0.5f * (pm[4] - pp[4]);
      // second-order stencils
      const float dyy = pc[7] - 2.0f * xc + pc[1];
      const float dss = pp[4] - 2.0f * xc + pm[4];
      const float dxy = 0.25f * (pc[0] - pc[2] - pc[6] + pc[8]);
      const float dys = 0.25f * (pp[1] - pp[7] - pm[1] + pm[7]);
      const float dxs = 0.25f * (pp[3] - pp[5] - pm[3] + pm[5]);

      // Hessian (faithful: H[2][2] duplicates dss) + constant jitter
      const float a00 = dss + JIT.v[0], a01 = dys + JIT.v[1], a02 = dxs + JIT.v[2];
      const float a10 = dys + JIT.v[3], a11 = dyy + JIT.v[4], a12 = dxy + JIT.v[5];
      const float a20 = dxs + JIT.v[6], a21 = dxy + JIT.v[7], a22 = dss + JIT.v[8];

      // 3x3 solve H*sol = [gz,gy,gx] via adjugate (Cramer); dx = -sol
      const float c00 = a11 * a22 - a12 * a21;
      const float c01 = a12 * a20 - a10 * a22;
      const float c02 = a10 * a21 - a11 * a20;
      const float det = a00 * c00 + a01 * c01 + a02 * c02;
      const float inv = 1.0f / det;
      const float s0 = (c00 * gz + (a02 * a21 - a01 * a22) * gy + (a01 * a12 - a02 * a11) * gx) * inv;
      const float s1 = (c01 * gz + (a00 * a22 - a02 * a20) * gy + (a02 * a10 - a00 * a12) * gx) * inv;
      const float s2 = (c02 * gz + (a01 * a20 - a00 * a21) * gy + (a00 * a11 - a01 * a10) * gx) * inv;
      const float d0 = -s0, d1 = -s1, d2 = -s2;

      // NMS: center equals 27-neighborhood max (== -inf-padded reduce_window)
      float m = xc;
#pragma unroll
      for (int k = 0; k < 9; ++k) {
        m = fmaxf(m, pm[k]); m = fmaxf(m, pc[k]); m = fmaxf(m, pp[k]);
      }
      const bool conv = (xc == m) &&
                        (fabsf(d0) < 0.5f) && (fabsf(d1) < 0.5f) && (fabsf(d2) < 0.5f);
      const float e0 = conv ? d0 : 0.0f;
      const float e1 = conv ? d1 : 0.0f;
      const float e2 = conv ? d2 : 0.0f;
      const float dy = 0.5f * (gz * e0 + gy * e1 + gx * e2);
      const float y  = xc + dy + (conv ? 10.0f : 0.0f);

      // ---- stores (streamed, nontemporal: 75 MB write-once) ----
      const int hg = h0 + hl, wg = w0 + wl;
      const size_t sidx = ((size_t)d * H_ + hg) * W_ + wg;
      float* cb = out + (size_t)b * 3 * NPB;
      __builtin_nontemporal_store(e0 + (float)d,  cb + 0 * NPB + sidx);
      __builtin_nontemporal_store(e1 + (float)hg, cb + 1 * NPB + sidx);
      __builtin_nontemporal_store(e2 + (float)wg, cb + 2 * NPB + sidx);
      __builtin_nontemporal_store(y, out + CO + (size_t)b * NPB + sidx);

      // roll the depth window: pm<-pc, pc<-pp, pp<-plane(min(d+2, D-1))
      if (d < D_ - 1) {
#pragma unroll
        for (int k = 0; k < 9; ++k) { pm[k] = pc[k]; pc[k] = pp[k]; }
        ldp(d + 2 < D_ ? d + 2 : D_ - 1, pp);
      }
    }
  }
}

extern "C" void kernel_launch(void* const* d_in, const int* in_sizes, int n_in,
                              void* d_out, int out_size, void* d_ws, size_t ws_size,
                              hipStream_t stream) {
  (void)in_sizes; (void)n_in; (void)out_size; (void)d_ws; (void)ws_size;
  const float* x = (const float*)d_in[0];
  float* out = (float*)d_out;
  dim3 grid(W_ / TW, H_ / TH, B_);   // (6, 48, 4)
  ConvQuadInterp3d_kernel<<<grid, dim3(256), 0, stream>>>(x, out);
}